// Predictor_28372553957803
// MI455X (gfx1250) — compile-verified
//
#include <hip/hip_runtime.h>
#include <hip/hip_bf16.h>

// ---------------------------------------------------------------------------
// CDNA5 (gfx1250) implementation of the sequential row-predictor scan.
// All conv layers run as implicit GEMM on the matrix pipe using
// V_WMMA_F32_16X16X4_F32 (fp32 in, fp32 accumulate -> bit-compatible with the
// fp32 reference, unlike f16/bf16 WMMA).
// ---------------------------------------------------------------------------

typedef float v2f __attribute__((ext_vector_type(2)));
typedef float v8f __attribute__((ext_vector_type(8)));

// ---------------------------------------------------------------------------
// One-time weight repack: w[co][ci][ky][kx] (3x3) -> wpk[co][(ci*3+ky)*4 + kx]
// with kx==3 padded to zero and co >= Cout padded to zero (so the WMMA loop
// is uniform even for the Cout=3 final layer). K' = Cin*12.
// ---------------------------------------------------------------------------
__global__ void repack_w(const float* __restrict__ w, float* __restrict__ wpk,
                         int Cin, int Cout) {
    int idx = blockIdx.x * 256 + threadIdx.x;
    int total = 64 * Cin * 12;
    if (idx >= total) return;
    int kp  = Cin * 12;
    int co  = idx / kp;
    int rem = idx - co * kp;
    int g   = rem >> 2;      // g = ci*3 + ky
    int kx  = rem & 3;
    float v = 0.f;
    if (co < Cout && kx < 3) {
        int ci = g / 3, ky = g - (g / 3) * 3;
        v = w[((co * Cin + ci) * 3 + ky) * 3 + kx];
    }
    wpk[idx] = v;
}

// ---------------------------------------------------------------------------
// 3x3 conv (stride 1, W-pad 1, H-pad ph) + bias + leaky ReLU via WMMA f32.
//
// GEMM view: out[n][co] = sum_k A[n][k] * B[k][co],  K' = Cin*12.
// A-fragment (16x4 f32, documented layout): lanes 0-15 hold K=0,1 (VGPR0/1),
// lanes 16-31 hold K=2,3  -> per lane two *contiguous* LDS floats
//   patch[(ci*3+ky)*72 + n + kxbase], kxbase = (lane>=16)?2:0.
// B-fragment (4x16 f32): lanes 0-15 hold K=k0,k0+1; lanes 16-31 K=k0+2,k0+3
//   -> per lane one aligned float2 from wpk[co*K' + g*4 + kxbase].
// D (16x16 f32, 8 VGPRs): VGPR r = spatial row r (lanes 0-15) / r+8 (16-31),
// N = lane&15 = co within tile.
//
// Workgroup: 256 threads (8 waves), one output row h, 64 columns.
// Waves 0-3: ntile 0-3, co-tiles {0,1}; waves 4-7: ntile 0-3, co-tiles {2,3}.
// ---------------------------------------------------------------------------
__global__ void __launch_bounds__(256)
conv3x3_wmma(const float* __restrict__ in, int inChanStride, int Hin,
             const float* __restrict__ wpk, const float* __restrict__ bias,
             float* __restrict__ out, int Cin, int Cout, int Hout, int ph) {
    __shared__ float patch[64 * 3 * 72];   // [ci*3+ky][72] (cols w0-1 .. w0+66)

    const int W   = 256;
    const int h   = blockIdx.y;
    const int w0  = blockIdx.x * 64;
    const int tid = threadIdx.x;

    // ---- stage input patch into LDS (zero-padded at H and W borders) ----
    const int nrows = Cin * 3;
    for (int idx = tid; idx < nrows * 68; idx += 256) {
        int row = idx / 68;            // ci*3 + ky
        int c   = idx - row * 68;      // 0..67  (input col = w0 + c - 1)
        int ci  = row / 3;
        int ky  = row - ci * 3;
        int hin = h + ky - ph;
        int win = w0 + c - 1;
        float v = 0.f;
        if ((unsigned)hin < (unsigned)Hin && (unsigned)win < (unsigned)W)
            v = in[ci * inChanStride + hin * W + win];
        patch[row * 72 + c] = v;
    }
    __syncthreads();

    const int lane    = tid & 31;
    const int wave    = tid >> 5;
    const int ntile   = wave & 3;
    const int cotile0 = (wave >> 2) * 2;          // 0 or 2
    const int Kp      = Cin * 12;

    const int m      = lane & 15;                 // M row / N col within tile
    const int kxbase = (lane >> 4) * 2;           // 0 (lanes 0-15) or 2
    const int aCol   = ntile * 16 + m + kxbase;   // LDS column for A.x

    v8f acc0 = {0.f, 0.f, 0.f, 0.f, 0.f, 0.f, 0.f, 0.f};
    v8f acc1 = {0.f, 0.f, 0.f, 0.f, 0.f, 0.f, 0.f, 0.f};

    const float* wrow0 = wpk + ( (size_t)(cotile0 * 16 + m) * Kp ) + kxbase;
    const float* wrow1 = wrow0 + (size_t)16 * Kp;

    const int G = Cin * 3;                        // k-steps of 4
    for (int g = 0; g < G; ++g) {
        v2f a;
        a.x = patch[g * 72 + aCol];
        a.y = patch[g * 72 + aCol + 1];
        v2f b0 = *(const v2f*)(wrow0 + g * 4);    // 8B aligned (Kp even, kxbase even)
        v2f b1 = *(const v2f*)(wrow1 + g * 4);
        acc0 = __builtin_amdgcn_wmma_f32_16x16x4_f32(
                   false, a, false, b0, (short)0, acc0, false, false);
        acc1 = __builtin_amdgcn_wmma_f32_16x16x4_f32(
                   false, a, false, b1, (short)0, acc1, false, false);
    }

    // ---- epilogue: bias + leaky ReLU + scattered store [co][h][w] ----
    const int co0  = cotile0 * 16 + m;
    const int co1  = co0 + 16;
    const float bi0 = (co0 < Cout) ? bias[co0] : 0.f;
    const float bi1 = (co1 < Cout) ? bias[co1] : 0.f;
    const int nbase = w0 + ntile * 16 + ((lane >> 4) ? 8 : 0);
#pragma unroll
    for (int r = 0; r < 8; ++r) {
        int n = nbase + r;
        float v0 = acc0[r] + bi0;
        v0 = (v0 >= 0.f) ? v0 : 0.01f * v0;
        if (co0 < Cout) out[(size_t)co0 * Hout * W + h * W + n] = v0;
        float v1 = acc1[r] + bi1;
        v1 = (v1 >= 0.f) ? v1 : 0.01f * v1;
        if (co1 < Cout) out[(size_t)co1 * Hout * W + h * W + n] = v1;
    }
}

// ---------------------------------------------------------------------------
// Per-pixel 8x8 @ 8x8:  p[i*8+k] = sum_j e[i*8+j] * s[j*8+k]
// Layouts all [64][9][256]. One thread per (h,w). 2.4 MFLOP total -> VALU.
// ---------------------------------------------------------------------------
__global__ void __launch_bounds__(256)
einsum8x8(const float* __restrict__ e, const float* __restrict__ s,
          float* __restrict__ out) {
    const int h  = blockIdx.x;
    const int w  = threadIdx.x;
    const int HS = 9 * 256;
    const int base = h * 256 + w;
    float sr[8][8];
#pragma unroll
    for (int j = 0; j < 8; ++j)
#pragma unroll
        for (int k = 0; k < 8; ++k)
            sr[j][k] = s[(j * 8 + k) * HS + base];
#pragma unroll
    for (int i = 0; i < 8; ++i) {
        float er[8];
#pragma unroll
        for (int j = 0; j < 8; ++j) er[j] = e[(i * 8 + j) * HS + base];
#pragma unroll
        for (int k = 0; k < 8; ++k) {
            float acc = 0.f;
#pragma unroll
            for (int j = 0; j < 8; ++j) acc += er[j] * sr[j][k];
            out[(i * 8 + k) * HS + base] = acc;
        }
    }
}

// ---------------------------------------------------------------------------
// Step update: truth = x[row i+15] - pr ; write ebuf row, pred slot, truth slot.
// d_out = [pred (3,16,256)][truth (3,16,256)].
// ---------------------------------------------------------------------------
__global__ void __launch_bounds__(256)
update_step(const float* __restrict__ x, const float* __restrict__ pr,
            float* __restrict__ ebuf, float* __restrict__ outp, int i) {
    int idx = blockIdx.x * 256 + threadIdx.x;   // 0..767
    int c = idx >> 8;
    int w = idx & 255;
    float p   = pr[c * 256 + w];
    float act = x[c * 31 * 256 + (i + 15) * 256 + w];
    float t   = act - p;
    ebuf[c * 31 * 256 + (i + 15) * 256 + w] = t;
    outp[c * 16 * 256 + i * 256 + w]              = p;
    outp[3 * 16 * 256 + c * 16 * 256 + i * 256 + w] = t;
}

// ---------------------------------------------------------------------------
// Host orchestration (graph-capture safe: only kernel launches + async d2d).
// ---------------------------------------------------------------------------
extern "C" void kernel_launch(void* const* d_in, const int* in_sizes, int n_in,
                              void* d_out, int out_size, void* d_ws, size_t ws_size,
                              hipStream_t stream) {
    (void)in_sizes; (void)n_in; (void)out_size; (void)ws_size;

    const float* x = (const float*)d_in[0];
    const float* w[13];
    const float* b[13];
    for (int j = 0; j < 13; ++j) {
        w[j] = (const float*)d_in[1 + 2 * j];
        b[j] = (const float*)d_in[2 + 2 * j];
    }
    static const int CIN[13]  = {3,64,64,64, 3,64,64,64, 64,64,64,64,64};
    static const int COUT[13] = {64,64,64,64,64,64,64,64,64,64,64,64, 3};

    // workspace carve-up (floats), 64-float (256 B) aligned blocks
    float* wsf = (float*)d_ws;
    size_t off = 0;
    auto alloc = [&](size_t n) {
        float* p = wsf + off;
        off += (n + 63) & ~(size_t)63;
        return p;
    };
    float* ebuf = alloc(3 * 31 * 256);
    float* bufA = alloc(64 * 15 * 256);
    float* bufB = alloc(64 * 15 * 256);
    float* bufC = alloc(64 * 15 * 256);
    float* bufD = alloc(64 * 15 * 256);
    float* bufP = alloc(64 * 9 * 256);
    float* bufQ = alloc(64 * 9 * 256);
    float* prb  = alloc(3 * 256);
    float* wpk[13];
    for (int j = 0; j < 13; ++j) wpk[j] = alloc((size_t)64 * CIN[j] * 12);

    // one-shot (per launch, deterministic) weight repack into WMMA-friendly K'
    for (int j = 0; j < 13; ++j) {
        int total = 64 * CIN[j] * 12;
        repack_w<<<(total + 255) / 256, 256, 0, stream>>>(w[j], wpk[j], CIN[j], COUT[j]);
    }
    // ebuf starts as a copy of x (must not mutate inputs)
    hipMemcpyAsync(ebuf, x, sizeof(float) * 3 * 31 * 256,
                   hipMemcpyDeviceToDevice, stream);

    auto conv = [&](const float* in, int chanStride, int Hin, int j,
                    float* out, int Hout, int ph) {
        dim3 grid(4, Hout);
        conv3x3_wmma<<<grid, 256, 0, stream>>>(in, chanStride, Hin, wpk[j], b[j],
                                               out, CIN[j], COUT[j], Hout, ph);
    };

    float* outp = (float*)d_out;
    for (int i = 0; i < 16; ++i) {
        // s-branch (weights *x) on ebuf rows i..i+14
        conv(ebuf + i * 256, 31 * 256, 15, 0, bufA, 15, 1);
        conv(bufA, 15 * 256, 15, 1, bufB, 13, 0);
        conv(bufB, 13 * 256, 13, 2, bufA, 11, 0);
        conv(bufA, 11 * 256, 11, 3, bufB, 9, 0);     // s result -> bufB
        // e-branch (weights *e) on the same slice
        conv(ebuf + i * 256, 31 * 256, 15, 4, bufC, 15, 1);
        conv(bufC, 15 * 256, 15, 5, bufD, 13, 0);
        conv(bufD, 13 * 256, 13, 6, bufC, 11, 0);
        conv(bufC, 11 * 256, 11, 7, bufD, 9, 0);     // e result -> bufD
        // per-pixel 8x8 matmul: pr = er @ sr
        einsum8x8<<<9, 256, 0, stream>>>(bufD, bufB, bufP);
        // tail convs 5..9
        conv(bufP, 9 * 256, 9, 8,  bufQ, 7, 0);
        conv(bufQ, 7 * 256, 7, 9,  bufP, 5, 0);
        conv(bufP, 5 * 256, 5, 10, bufQ, 3, 0);
        conv(bufQ, 3 * 256, 3, 11, bufP, 1, 0);
        conv(bufP, 1 * 256, 1, 12, prb,  1, 1);      // Cout = 3
        // scan update + emit pred/truth rows
        update_step<<<3, 256, 0, stream>>>(x, prb, ebuf, outp, i);
    }
}